// AdvancedTemporalTransactionGNN_2774548873491
// MI455X (gfx1250) — compile-verified
//
#include <hip/hip_runtime.h>
#include <hip/hip_bf16.h>
#include <math.h>

typedef __attribute__((ext_vector_type(2))) float v2f;
typedef __attribute__((ext_vector_type(8))) float v8f;

#define DIM 128
#define HEADS 4
#define HEAD_DIM 32

// ---------------------------------------------------------------------------
// WMMA fp32 GEMM: Y[M,128] = X[M,128] @ W[128,128] + bias[128]
// One wave computes one 16x16 tile of Y, looping K in steps of 4 with
// v_wmma_f32_16x16x4_f32 (full fp32 precision, matches reference numerics).
// A 16x4 layout : lane l, vgpr j -> A[M = l%16][K = kb + 2*(l>>4) + j]
// B 4x16 layout : lane l, vgpr j -> B[K = kb + 2*(l>>4) + j][N = l%16]
// C 16x16 layout: lane l, vgpr r -> C[M = r + 8*(l>>4)][N = l%16]
// ---------------------------------------------------------------------------
__global__ __launch_bounds__(256) void wmma_gemm_bias(
    const float* __restrict__ X, const float* __restrict__ W,
    const float* __restrict__ bias, float* __restrict__ Y,
    int rowTiles, int M)
{
    const int lane   = threadIdx.x & 31;
    const int waveIB = threadIdx.x >> 5;
    const long wave  = (long)blockIdx.x * (blockDim.x >> 5) + waveIB;
    if (wave >= (long)rowTiles * 8) return;   // wave-uniform: EXEC stays full

    const int rowTile = (int)(wave >> 3);
    const int colTile = (int)(wave & 7);
    const int rowBase = rowTile * 16;
    const int colBase = colTile * 16;
    const int laneLo  = lane & 15;
    const int laneHi  = lane >> 4;            // 0 or 1
    const int col     = colBase + laneLo;

    int mRow = rowBase + laneLo;
    if (mRow > M - 1) mRow = M - 1;           // clamp (branchless, keeps EXEC)
    const float* xrow = X + (long)mRow * DIM;

    v8f c = {};
    #pragma unroll 4
    for (int kb = 0; kb < DIM; kb += 4) {
        const int k0 = kb + 2 * laneHi;       // even -> 8B aligned
        v2f a = *(const v2f*)(xrow + k0);
        v2f b;
        b[0] = W[(long)k0 * DIM + col];
        b[1] = W[(long)(k0 + 1) * DIM + col];
        c = __builtin_amdgcn_wmma_f32_16x16x4_f32(
                false, a, false, b, (short)0, c, false, false);
    }

    const float bc = bias[col];
    if (rowBase + 16 <= M) {
        // Wave-uniform fast path (always taken when M % 16 == 0): straight
        // stores, no exec-mask manipulation in the epilogue.
        #pragma unroll
        for (int r = 0; r < 8; ++r) {
            const int row = rowBase + r + 8 * laneHi;
            Y[(long)row * DIM + col] = c[r] + bc;
        }
    } else {
        #pragma unroll
        for (int r = 0; r < 8; ++r) {
            const int row = rowBase + r + 8 * laneHi;
            if (row < M) Y[(long)row * DIM + col] = c[r] + bc;
        }
    }
}

// ---------------------------------------------------------------------------
// Helpers: monotone float<->uint mapping so unsigned atomicMax == float max
// ---------------------------------------------------------------------------
__device__ __forceinline__ unsigned fmap(float f) {
    unsigned b = __float_as_uint(f);
    return (b & 0x80000000u) ? ~b : (b | 0x80000000u);
}
__device__ __forceinline__ float funmap(unsigned u) {
    unsigned b = (u & 0x80000000u) ? (u & 0x7FFFFFFFu) : ~u;
    return __uint_as_float(b);
}

__global__ void init_stats(unsigned* maxbits, float* sumv) {
    if (threadIdx.x < HEADS) { maxbits[threadIdx.x] = 0u; sumv[threadIdx.x] = 0.0f; }
}

__global__ void zero_f4(float4* p, long n4) {
    long i = (long)blockIdx.x * blockDim.x + threadIdx.x;
    if (i < n4) p[i] = make_float4(0.f, 0.f, 0.f, 0.f);
}

// ---------------------------------------------------------------------------
// Per-(edge,head) attention score + temporal bias; global per-head max.
// ---------------------------------------------------------------------------
__global__ __launch_bounds__(256) void edge_scores(
    const float* __restrict__ q, const float* __restrict__ k,
    const int* __restrict__ src, const int* __restrict__ dst,
    const float* __restrict__ edge_time, const float* __restrict__ node_time,
    const float* __restrict__ Wt, const float* __restrict__ bt,
    float* __restrict__ scores, unsigned* __restrict__ maxbits, int E)
{
    __shared__ unsigned smax[HEADS];
    if (threadIdx.x < HEADS) smax[threadIdx.x] = 0u;
    __syncthreads();

    const long idx = (long)blockIdx.x * blockDim.x + threadIdx.x;
    const bool active = idx < (long)E * HEADS;
    if (active) {
        const int e = (int)(idx >> 2);
        const int h = (int)(idx & 3);
        const int s = src[e], d = dst[e];
        const float4* qp = (const float4*)(q + (long)d * DIM + h * HEAD_DIM);
        const float4* kp = (const float4*)(k + (long)s * DIM + h * HEAD_DIM);
        float dot = 0.0f;
        #pragma unroll
        for (int i = 0; i < HEAD_DIM / 4; ++i) {
            float4 a = qp[i], b = kp[i];
            dot += a.x * b.x + a.y * b.y + a.z * b.z + a.w * b.w;
        }
        const float td = edge_time[e] - node_time[d];
        const float f0 = (td > 0.0f) ? 1.0f : ((td < 0.0f) ? -1.0f : 0.0f);
        const float f1 = log1pf(fabsf(td) * (1.0f / 3600.0f));
        const float tw = f0 * Wt[h] + f1 * Wt[HEADS + h] + bt[h];
        const float score = dot * 0.17677669529663687f /* 32^-0.5 */ + tw;
        scores[idx] = score;
        atomicMax(&smax[h], fmap(score));
    }
    __syncthreads();
    if (threadIdx.x < HEADS) atomicMax(&maxbits[threadIdx.x], smax[threadIdx.x]);
}

// exp(score - max) written in place; global per-head sum.
__global__ __launch_bounds__(256) void sum_exp(
    float* __restrict__ scores, const unsigned* __restrict__ maxbits,
    float* __restrict__ sumv, int E)
{
    __shared__ float ssum[HEADS];
    if (threadIdx.x < HEADS) ssum[threadIdx.x] = 0.0f;
    __syncthreads();

    const long idx = (long)blockIdx.x * blockDim.x + threadIdx.x;
    if (idx < (long)E * HEADS) {
        const int h = (int)(idx & 3);
        const float mx = funmap(maxbits[h]);
        const float ex = expf(scores[idx] - mx);
        scores[idx] = ex;
        atomicAdd(&ssum[h], ex);
    }
    __syncthreads();
    if (threadIdx.x < HEADS) atomicAdd(&sumv[threadIdx.x], ssum[threadIdx.x]);
}

// One thread per (edge, feature): coalesced v[src] read, atomic scatter to dst.
__global__ __launch_bounds__(256) void scatter_messages(
    const float* __restrict__ v, const float* __restrict__ scores,
    const float* __restrict__ sumv,
    const int* __restrict__ src, const int* __restrict__ dst,
    float* __restrict__ accum, int E)
{
    const long idx = (long)blockIdx.x * blockDim.x + threadIdx.x;
    if (idx >= (long)E * DIM) return;
    const int e = (int)(idx >> 7);
    const int f = (int)(idx & 127);
    const int h = f >> 5;
    const float alpha = scores[(long)e * HEADS + h] / sumv[h];
    const float msg = alpha * v[(long)src[e] * DIM + f];
    atomicAdd(&accum[(long)dst[e] * DIM + f], msg);
}

// ---------------------------------------------------------------------------
extern "C" void kernel_launch(void* const* d_in, const int* in_sizes, int n_in,
                              void* d_out, int out_size, void* d_ws, size_t ws_size,
                              hipStream_t stream)
{
    const float* x         = (const float*)d_in[0];
    const int*   eidx      = (const int*)  d_in[1];
    const float* edge_time = (const float*)d_in[2];
    const float* node_time = (const float*)d_in[3];
    const float* Wq = (const float*)d_in[4];  const float* bq = (const float*)d_in[5];
    const float* Wk = (const float*)d_in[6];  const float* bk = (const float*)d_in[7];
    const float* Wv = (const float*)d_in[8];  const float* bv = (const float*)d_in[9];
    const float* Wt = (const float*)d_in[10]; const float* bt = (const float*)d_in[11];
    const float* Wo = (const float*)d_in[12]; const float* bo = (const float*)d_in[13];
    float* out = (float*)d_out;

    const int N = in_sizes[0] / DIM;
    const int E = in_sizes[2];
    const int* src = eidx;
    const int* dst = eidx + E;

    // Workspace layout (floats): q | k | v | scores(E*4) | maxbits(4) | sumv(4)
    float*    q       = (float*)d_ws;
    float*    kbuf    = q    + (size_t)N * DIM;
    float*    vbuf    = kbuf + (size_t)N * DIM;
    float*    scores  = vbuf + (size_t)N * DIM;
    unsigned* maxbits = (unsigned*)(scores + (size_t)E * HEADS);
    float*    sumv    = (float*)(maxbits + HEADS);
    float*    accum   = q;   // q is dead after edge_scores -> reuse as accumulator

    const int rowTiles   = (N + 15) / 16;
    const int gemmBlocks = (rowTiles * 8 + 7) / 8;          // 8 waves / 256-thr block
    const long eh        = (long)E * HEADS;
    const int  ehBlocks  = (int)((eh + 255) / 256);
    const long ef        = (long)E * DIM;
    const int  efBlocks  = (int)((ef + 255) / 256);
    const long n4        = (long)N * DIM / 4;
    const int  zBlocks   = (int)((n4 + 255) / 256);

    // 1) q/k/v projections (WMMA fp32)
    wmma_gemm_bias<<<gemmBlocks, 256, 0, stream>>>(x, Wq, bq, q,    rowTiles, N);
    wmma_gemm_bias<<<gemmBlocks, 256, 0, stream>>>(x, Wk, bk, kbuf, rowTiles, N);
    wmma_gemm_bias<<<gemmBlocks, 256, 0, stream>>>(x, Wv, bv, vbuf, rowTiles, N);

    // 2) softmax statistics + scores
    init_stats<<<1, 32, 0, stream>>>(maxbits, sumv);
    edge_scores<<<ehBlocks, 256, 0, stream>>>(q, kbuf, src, dst, edge_time,
                                              node_time, Wt, bt, scores, maxbits, E);

    // 3) zero the accumulator (reuses q), then exp + global sum
    zero_f4<<<zBlocks, 256, 0, stream>>>((float4*)accum, n4);
    sum_exp<<<ehBlocks, 256, 0, stream>>>(scores, maxbits, sumv, E);

    // 4) weighted message scatter-add
    scatter_messages<<<efBlocks, 256, 0, stream>>>(vbuf, scores, sumv, src, dst,
                                                   accum, E);

    // 5) output projection (WMMA fp32)
    wmma_gemm_bias<<<gemmBlocks, 256, 0, stream>>>(accum, Wo, bo, out, rowTiles, N);
}